// Extractor_85933705658892
// MI455X (gfx1250) — compile-verified
//
#include <hip/hip_runtime.h>
#include <math.h>

typedef __attribute__((ext_vector_type(2))) float v2f;
typedef __attribute__((ext_vector_type(8))) float v8f;

#define N_WIN   25
#define C_IN    7
#define L_IN    3000
#define C_OUT   64
#define KSZ     100
#define STRIDE  20
#define L_OUT   146            // (3000-100)/20 + 1
#define CO_TILES 4             // 64 / 16
#define L_TILES  10            // ceil(146/16)
#define TILES_TOTAL (N_WIN * CO_TILES * L_TILES)   // 1000
#define WAVES_PER_BLOCK 8

// GEMM-as-conv: D(16x16) = A(16x4 of W) x B(4x16 of x-patches) + C, f32 exact.
// A-matrix (16x4 f32): lanes 0-15 hold M=0..15; VGPR0 = K {0,2} by lane half,
// VGPR1 = K {1,3}. B-matrix (4x16 f32) mirrors: VGPR0 = rows K {0,2}, VGPR1 =
// rows K {1,3}, columns striped over lanes. So each lane loads a contiguous,
// 8B-aligned float2 from both W and x per WMMA.
__global__ __launch_bounds__(256) void conv_wmma_kernel(
    const float* __restrict__ x,   // [25, 7, 3000]
    const float* __restrict__ W,   // [64, 7, 100]
    float* __restrict__ lin)       // [25, 64, 146]
{
    const int wave = threadIdx.x >> 5;
    const int lane = threadIdx.x & 31;
    const int tile = blockIdx.x * WAVES_PER_BLOCK + wave;   // grid sized exactly

    const int lt = tile % L_TILES;
    const int ct = (tile / L_TILES) % CO_TILES;
    const int n  = tile / (L_TILES * CO_TILES);

    const int half = lane >> 4;    // which K-pair this lane holds
    const int lid  = lane & 15;

    const int col  = lt * 16 + lid;                         // output position
    const int posc = (col < L_OUT - 1) ? col : (L_OUT - 1); // clamp for loads
    const int arow = ct * 16 + lid;                         // c_out row

    const float* wp_base = W + arow * (C_IN * KSZ) + 2 * half;
    const float* xp_base = x + n * (C_IN * L_IN) + posc * STRIDE + 2 * half;

    v8f acc = {0.f, 0.f, 0.f, 0.f, 0.f, 0.f, 0.f, 0.f};

    // K = 700 = 7 channels x 100 taps; 100 % 4 == 0 so a K=4 group never
    // crosses a channel boundary.
    for (int ci = 0; ci < C_IN; ++ci) {
        const float* wp = wp_base + ci * KSZ;
        const float* xp = xp_base + ci * L_IN;
#pragma unroll
        for (int kk = 0; kk < KSZ; kk += 4) {
            v2f a = *(const v2f*)(wp + kk);   // W[arow, ci, kk + 2*half + {0,1}]
            v2f b = *(const v2f*)(xp + kk);   // x[n, ci, posc*20 + kk + 2*half + {0,1}]
            acc = __builtin_amdgcn_wmma_f32_16x16x4_f32(
                /*neg_a=*/false, a, /*neg_b=*/false, b,
                /*c_mod=*/(short)0, acc, /*reuse_a=*/false, /*reuse_b=*/false);
        }
    }

    // C/D layout: VGPR r -> row M = r + 8*half, column N = lid. Divergent
    // masking only here, after all WMMAs (EXEC all-ones during WMMA).
    if (col < L_OUT) {
        float* outp = lin + n * (C_OUT * L_OUT) + (ct * 16 + 8 * half) * L_OUT + col;
#pragma unroll
        for (int r = 0; r < 8; ++r)
            outp[r * L_OUT] = acc[r];
    }
}

// res_t = tanh(lin_t + res_{t-1}), res_0 = 0. Each thread owns one of the
// 64*146 columns across all 25 steps -> in-place update of d_out is safe.
__global__ void recur_kernel(float* __restrict__ out)
{
    const int i = blockIdx.x * blockDim.x + threadIdx.x;
    const int plane = C_OUT * L_OUT;   // 9344
    if (i >= plane) return;
    float res = 0.f;
#pragma unroll
    for (int t = 0; t < N_WIN; ++t) {
        res = tanhf(out[t * plane + i] + res);
        out[t * plane + i] = res;
    }
}

extern "C" void kernel_launch(void* const* d_in, const int* in_sizes, int n_in,
                              void* d_out, int out_size, void* d_ws, size_t ws_size,
                              hipStream_t stream)
{
    (void)in_sizes; (void)n_in; (void)d_ws; (void)ws_size; (void)out_size;
    const float* x = (const float*)d_in[0];   // [25,7,3000]
    const float* W = (const float*)d_in[1];   // [64,7,100]
    float* out = (float*)d_out;               // [25,64,146]

    conv_wmma_kernel<<<TILES_TOTAL / WAVES_PER_BLOCK, 256, 0, stream>>>(x, W, out);

    const int plane = C_OUT * L_OUT;
    recur_kernel<<<(plane + 255) / 256, 256, 0, stream>>>(out);
}